// ScaledDotProductAttention_28673201668215
// MI455X (gfx1250) — compile-verified
//
#include <hip/hip_runtime.h>
#include <math.h>

typedef __attribute__((ext_vector_type(2))) float v2f;
typedef __attribute__((ext_vector_type(8))) float v8f;

#define SEQ 2048
#define DK 64
#define NEG_INF_F (-1.0e9f)

__device__ __forceinline__ float rowmax16(float v) {
    v = fmaxf(v, __shfl_xor(v, 1, 32));
    v = fmaxf(v, __shfl_xor(v, 2, 32));
    v = fmaxf(v, __shfl_xor(v, 4, 32));
    v = fmaxf(v, __shfl_xor(v, 8, 32));
    return v;
}
__device__ __forceinline__ float rowsum16(float v) {
    v += __shfl_xor(v, 1, 32);
    v += __shfl_xor(v, 2, 32);
    v += __shfl_xor(v, 4, 32);
    v += __shfl_xor(v, 8, 32);
    return v;
}

// One wave = one 16-query tile of one (b,h). 4 waves per block.
// Two-pass flash-style softmax: pass1 computes per-row (m, l) online;
// pass2 recomputes scores, writes attn (non-temporal) and accumulates P@V.
__global__ __launch_bounds__(128) void sdpa_wmma_f32_kernel(
    const float* __restrict__ Q, const float* __restrict__ K,
    const float* __restrict__ V, const unsigned char* __restrict__ mask,
    float* __restrict__ prob, float* __restrict__ attn)
{
    __shared__ float pbuf[4][16 * 18];   // per-wave 16x16 P tile, stride 18 (bank-conflict pad, keeps 8B align)

    const int lane = threadIdx.x & 31;
    const int wave = threadIdx.x >> 5;
    const int hi   = lane >> 4;          // half-wave select (0/1)
    const int ln   = lane & 15;

    const int qTilesPerBH = SEQ / 64;    // 32 (block covers 64 queries)
    const int bh   = blockIdx.x / qTilesPerBH;
    const int qblk = blockIdx.x % qTilesPerBH;
    const int q0   = qblk * 64 + wave * 16;

    const float* __restrict__ Qb = Q + (size_t)bh * SEQ * DK;
    const float* __restrict__ Kb = K + (size_t)bh * SEQ * DK;
    const float* __restrict__ Vb = V + (size_t)bh * SEQ * DK;
    const unsigned char* __restrict__ Mb = mask + (size_t)bh * SEQ * SEQ;
    float* __restrict__ Pout = prob + (size_t)bh * SEQ * DK;
    float* __restrict__ Aout = attn + (size_t)bh * SEQ * SEQ;

    const float scale = 0.125f;          // 1/sqrt(64)

    // ---- Cache Q tile (16x64) as WMMA A-fragments: lane<16 -> {K0,K1}, lane>=16 -> {K2,K3}
    v2f aq[16];
    {
        const size_t qrow = (size_t)(q0 + ln) * DK;
        #pragma unroll
        for (int d = 0; d < 16; ++d) {
            const int col = d * 4 + 2 * hi;
            aq[d] = *reinterpret_cast<const v2f*>(Qb + qrow + col);
        }
    }

    float m_run[8], l_run[8];
    #pragma unroll
    for (int r = 0; r < 8; ++r) { m_run[r] = -3.0e38f; l_run[r] = 0.0f; }

    // ================= Pass 1: online softmax stats (m, l) per row =================
    for (int k0 = 0; k0 < SEQ; k0 += 16) {
        v8f c = {};
        const size_t krow = (size_t)(k0 + ln) * DK;
        #pragma unroll
        for (int d = 0; d < 16; ++d) {
            const int col = d * 4 + 2 * hi;
            v2f bk = *reinterpret_cast<const v2f*>(Kb + krow + col);
            c = __builtin_amdgcn_wmma_f32_16x16x4_f32(false, aq[d], false, bk,
                                                      (short)0, c, false, false);
        }
        #pragma unroll
        for (int r = 0; r < 8; ++r) {
            const int qrow = q0 + r + 8 * hi;
            const unsigned char mk = Mb[(size_t)qrow * SEQ + k0 + ln];
            float s = mk ? NEG_INF_F : c[r] * scale;
            float tmax = rowmax16(s);
            float mnew = fmaxf(m_run[r], tmax);
            float corr = __expf(m_run[r] - mnew);
            float p    = __expf(s - mnew);
            float rsum = rowsum16(p);
            l_run[r] = l_run[r] * corr + rsum;
            m_run[r] = mnew;
        }
    }

    float inv_l[8];
    #pragma unroll
    for (int r = 0; r < 8; ++r) inv_l[r] = 1.0f / l_run[r];

    // ================= Pass 2: write attn, accumulate O = P @ V =================
    v8f o0 = {}, o1 = {}, o2 = {}, o3 = {};
    float* __restrict__ myP = pbuf[wave];

    for (int k0 = 0; k0 < SEQ; k0 += 16) {
        v8f c = {};
        const size_t krow = (size_t)(k0 + ln) * DK;
        #pragma unroll
        for (int d = 0; d < 16; ++d) {
            const int col = d * 4 + 2 * hi;
            v2f bk = *reinterpret_cast<const v2f*>(Kb + krow + col);
            c = __builtin_amdgcn_wmma_f32_16x16x4_f32(false, aq[d], false, bk,
                                                      (short)0, c, false, false);
        }
        // Normalize, stream attn out (NT: 537MB, don't pollute L2), stage P in LDS
        #pragma unroll
        for (int r = 0; r < 8; ++r) {
            const int qrow = q0 + r + 8 * hi;
            const unsigned char mk = Mb[(size_t)qrow * SEQ + k0 + ln];
            float s = mk ? NEG_INF_F : c[r] * scale;
            float p = __expf(s - m_run[r]) * inv_l[r];
            __builtin_nontemporal_store(p, Aout + (size_t)qrow * SEQ + k0 + ln);
            myP[(r + 8 * hi) * 18 + ln] = p;
        }
        // O += P(16x16) @ V(16x64): A-frags of P from LDS, B-frags of V from global
        #pragma unroll
        for (int kk = 0; kk < 4; ++kk) {
            const int col = kk * 4 + 2 * hi;
            v2f ap = *reinterpret_cast<const v2f*>(myP + ln * 18 + col);
            const float* vr0 = Vb + (size_t)(k0 + col)     * DK;  // key for .x
            const float* vr1 = Vb + (size_t)(k0 + col + 1) * DK;  // key for .y
            {
                v2f bv; bv.x = vr0[ 0 + ln]; bv.y = vr1[ 0 + ln];
                o0 = __builtin_amdgcn_wmma_f32_16x16x4_f32(false, ap, false, bv, (short)0, o0, false, false);
            }
            {
                v2f bv; bv.x = vr0[16 + ln]; bv.y = vr1[16 + ln];
                o1 = __builtin_amdgcn_wmma_f32_16x16x4_f32(false, ap, false, bv, (short)0, o1, false, false);
            }
            {
                v2f bv; bv.x = vr0[32 + ln]; bv.y = vr1[32 + ln];
                o2 = __builtin_amdgcn_wmma_f32_16x16x4_f32(false, ap, false, bv, (short)0, o2, false, false);
            }
            {
                v2f bv; bv.x = vr0[48 + ln]; bv.y = vr1[48 + ln];
                o3 = __builtin_amdgcn_wmma_f32_16x16x4_f32(false, ap, false, bv, (short)0, o3, false, false);
            }
        }
    }

    // ---- Write prob (O) tile: row = r + 8*hi, cols split over 4 tiles of 16
    #pragma unroll
    for (int r = 0; r < 8; ++r) {
        float* orow = Pout + (size_t)(q0 + r + 8 * hi) * DK;
        orow[ 0 + ln] = o0[r];
        orow[16 + ln] = o1[r];
        orow[32 + ln] = o2[r];
        orow[48 + ln] = o3[r];
    }
}

extern "C" void kernel_launch(void* const* d_in, const int* in_sizes, int n_in,
                              void* d_out, int out_size, void* d_ws, size_t ws_size,
                              hipStream_t stream) {
    const float*         Q    = (const float*)d_in[0];
    const float*         K    = (const float*)d_in[1];
    const float*         V    = (const float*)d_in[2];
    const unsigned char* mask = (const unsigned char*)d_in[3];  // jnp.bool_ -> 1 byte/elem

    const int n_bh = in_sizes[0] / (SEQ * DK);   // B*H = 32

    float* out  = (float*)d_out;
    float* prob = out;                                   // [B,H,S,D] first
    float* attn = out + (size_t)n_bh * SEQ * DK;         // then [B,H,S,S]

    dim3 grid(n_bh * (SEQ / 64));   // 1024 blocks
    dim3 block(128);                // 4 waves, one 16-query tile each
    sdpa_wmma_f32_kernel<<<grid, block, 0, stream>>>(Q, K, V, mask, prob, attn);
}